// Stage2FairnessLoss_63230508532319
// MI455X (gfx1250) — compile-verified
//
#include <hip/hip_runtime.h>
#include <math.h>

// ---------------------------------------------------------------------------
// Sinkhorn fairness loss for MI455X (gfx1250, wave32).
//  Phase 1: 128 fp32 GEMMs via V_WMMA_F32_16X16X4_F32 (fp32 accuracy is
//           mandatory: the last softmin divides by eps=1e-8).
//  Phase 2: 142 softmin sweeps, transcendental-bound; C matrices (~128MB)
//           stay resident in the 192MB L2 between sweeps. Inner loop uses
//           b128 global/LDS accesses to minimize issue-slot pressure.
// ---------------------------------------------------------------------------

typedef __attribute__((ext_vector_type(2))) float v2f;
typedef __attribute__((ext_vector_type(4))) float v4f;
typedef __attribute__((ext_vector_type(8))) float v8f;

#define DIMF   512
#define NPER   512
#define NPAIRS 56
#define NSELF  16
#define MATSZ  (NPER * NPER)      // 262144 floats per cost matrix
#define LOG_N  6.23832462503951f  // log(512)
#define EPS_F  1e-8f              // BLUR^P

// triu_indices(8, k=1) -> 28 pairs
__constant__ int c_pi[28] = {0,0,0,0,0,0,0,1,1,1,1,1,1,2,2,2,2,2,3,3,3,3,4,4,4,5,5,6};
__constant__ int c_pj[28] = {1,2,3,4,5,6,7,2,3,4,5,6,7,3,4,5,6,7,4,5,6,7,5,6,7,6,7,7};

__device__ __forceinline__ float wred_max(float v) {
    #pragma unroll
    for (int o = 16; o > 0; o >>= 1) v = fmaxf(v, __shfl_xor(v, o, 32));
    return v;
}
__device__ __forceinline__ float wred_sum(float v) {
    #pragma unroll
    for (int o = 16; o > 0; o >>= 1) v += __shfl_xor(v, o, 32);
    return v;
}

// ---- gather into balanced (class,subgroup) buckets -------------------------
// stable argsort of labels*8+subgroups with labels=i/4096, subgroups=i%8
// => bucket (c,s) element k = row c*4096 + s + 8*k
__global__ void k_gather(const float* __restrict__ f, float* __restrict__ out) {
    unsigned idx = blockIdx.x * blockDim.x + threadIdx.x;  // 16*512*512 total
    int d = idx & 511;
    int k = (idx >> 9) & 511;
    int b = idx >> 18;                 // bucket 0..15
    int c = b >> 3, s = b & 7;
    out[idx] = f[(size_t)(c * 4096 + s + 8 * k) * DIMF + d];
}

// ---- squared row norms (one wave per row) ---------------------------------
__global__ void k_norms(const float* __restrict__ fs, float* __restrict__ n2) {
    int wave = (blockIdx.x * blockDim.x + threadIdx.x) >> 5;
    int lane = threadIdx.x & 31;
    const v4f* row = (const v4f*)(fs + (size_t)wave * DIMF);
    float s = 0.f;
    #pragma unroll
    for (int j = 0; j < 4; j++) {
        v4f v = row[lane + 32 * j];
        s += v.x * v.x + v.y * v.y + v.z * v.z + v.w * v.w;
    }
    s = wred_sum(s);
    if (lane == 0) n2[wave] = s;
}

// ---- per-pair eps0 = diameter^2 = sum_d (max_d - min_d)^2 -----------------
__global__ void k_eps0(const float* __restrict__ fs, float* __restrict__ eps0) {
    int p = blockIdx.x;                 // 0..55
    int c = p / 28, t = p % 28;
    const float* X = fs + (size_t)(c * 8 + c_pi[t]) * NPER * DIMF;
    const float* Y = fs + (size_t)(c * 8 + c_pj[t]) * NPER * DIMF;
    float acc = 0.f;
    for (int d = threadIdx.x; d < DIMF; d += blockDim.x) {
        float mx = -3.0e38f, mn = 3.0e38f;
        for (int k = 0; k < NPER; k++) {
            float v = X[(size_t)k * DIMF + d];
            mx = fmaxf(mx, v); mn = fminf(mn, v);
        }
        for (int k = 0; k < NPER; k++) {
            float v = Y[(size_t)k * DIMF + d];
            mx = fmaxf(mx, v); mn = fminf(mn, v);
        }
        float r = mx - mn;
        acc += r * r;
    }
    __shared__ float red[256];
    red[threadIdx.x] = acc;
    __syncthreads();
    for (int o = 128; o > 0; o >>= 1) {
        if (threadIdx.x < (unsigned)o) red[threadIdx.x] += red[threadIdx.x + o];
        __syncthreads();
    }
    if (threadIdx.x == 0) eps0[p] = red[0];
}

// ---- cost matrices via fp32 WMMA ------------------------------------------
// blockIdx.y = matrix id: [0,56) Cxy, [56,112) Cyx, [112,128) Cself
// block = 256 threads = 8 waves, each wave one 16x16 output tile.
__global__ void k_gemm(const float* __restrict__ fs, const float* __restrict__ n2,
                       float* __restrict__ Cxy, float* __restrict__ Cyx,
                       float* __restrict__ Cself) {
    int mat  = blockIdx.y;
    int wave = threadIdx.x >> 5, lane = threadIdx.x & 31;
    int tile = blockIdx.x * 8 + wave;        // 0..1023 (32x32 tiles)
    int tm = (tile >> 5) << 4;
    int tn = (tile & 31) << 4;

    int ab, bb; float* dst;
    if (mat < 56) {
        int c = mat / 28, t = mat % 28;
        ab = c * 8 + c_pi[t]; bb = c * 8 + c_pj[t];
        dst = Cxy + (size_t)mat * MATSZ;
    } else if (mat < 112) {
        int p = mat - 56; int c = p / 28, t = p % 28;
        ab = c * 8 + c_pj[t]; bb = c * 8 + c_pi[t];
        dst = Cyx + (size_t)p * MATSZ;
    } else {
        int s = mat - 112; ab = bb = s;
        dst = Cself + (size_t)s * MATSZ;
    }
    const float* A = fs + (size_t)ab * NPER * DIMF;
    const float* B = fs + (size_t)bb * NPER * DIMF;

    int half = lane >> 4, r = lane & 15;
    v8f acc = {0.f, 0.f, 0.f, 0.f, 0.f, 0.f, 0.f, 0.f};

    // A: 16x4 fp32 (lane r holds row tm+r, K = 2*half..2*half+1)
    // B: 4x16 fp32 (lane r holds col tn+r of Y, same K split) -> C = A*B = X.Y^T
    const float* ap = A + (size_t)(tm + r) * DIMF + 2 * half;
    const float* bp = B + (size_t)(tn + r) * DIMF + 2 * half;
    #pragma unroll 8
    for (int k = 0; k < DIMF; k += 4) {
        v2f a = *(const v2f*)(ap + k);
        v2f b = *(const v2f*)(bp + k);
        acc = __builtin_amdgcn_wmma_f32_16x16x4_f32(false, a, false, b,
                                                    (short)0, acc, false, false);
    }

    // C/D layout: lanes 0-15: N=lane, M=vgpr; lanes 16-31: N=lane-16, M=vgpr+8
    const float* n2a = n2 + ab * NPER;
    const float* n2b = n2 + bb * NPER;
    #pragma unroll
    for (int v = 0; v < 8; v++) {
        int M = v + 8 * half;
        float cost = fmaxf(0.5f * (n2a[tm + M] + n2b[tn + r]) - acc[v], 0.0f);
        dst[(size_t)(tm + M) * NPER + tn + r] = cost;
    }
}

// ---- softmin sweep ---------------------------------------------------------
// grid.x = 56 pairs * 4 matrices * 64 chunks; block = 256 = 8 waves,
// one wave per row (512 cols, 16 per lane via 4x b128), g staged in LDS.
// mat: 0=xy(out f_ba, g=g_ab) 1=yx(out g_ab, g=f_ba) 2=xx(f_aa) 3=yy(g_bb)
__global__ void k_softmin(const float* __restrict__ Cxy, const float* __restrict__ Cyx,
                          const float* __restrict__ Cself, const float* __restrict__ eps0,
                          const float* __restrict__ potIn, float* __restrict__ potOut,
                          float epsScale, float mixOld, float mixNew,
                          int finalMode, float* __restrict__ out) {
    int bid   = blockIdx.x;
    int chunk = bid & 63;
    int mat   = (bid >> 6) & 3;
    int pair  = bid >> 8;
    int wave  = threadIdx.x >> 5, lane = threadIdx.x & 31;
    int row   = chunk * 8 + wave;

    int c = pair / 28, t = pair % 28;
    const float* Cp;
    if (mat == 0)      Cp = Cxy + (size_t)pair * MATSZ;
    else if (mat == 1) Cp = Cyx + (size_t)pair * MATSZ;
    else if (mat == 2) Cp = Cself + (size_t)(c * 8 + c_pi[t]) * MATSZ;
    else               Cp = Cself + (size_t)(c * 8 + c_pj[t]) * MATSZ;

    int gidx = (mat == 0) ? 1 : (mat == 1) ? 0 : mat;   // dual potential index
    const float* gsrc = potIn + ((size_t)pair * 4 + gidx) * NPER;

    const v4f* Crow4 = (const v4f*)(Cp + (size_t)row * NPER);
    __builtin_prefetch(Crow4 + lane, 0, 1);            // global_prefetch_b8

    __shared__ __align__(16) float g[NPER];
    for (int i = threadIdx.x; i < NPER; i += 256) g[i] = gsrc[i];
    __syncthreads();

    float eps = finalMode ? EPS_F : fmaxf(eps0[pair] * epsScale, EPS_F);
    float inv_eps = 1.0f / eps;

    const v4f* g4 = (const v4f*)g;
    v4f a4[4];
    float amax = -3.0e38f;
    #pragma unroll
    for (int j = 0; j < 4; j++) {
        int idx = lane + 32 * j;                       // 512B-contig per wave
        v4f cv = Crow4[idx];
        v4f gv = g4[idx];
        v4f v = (gv - cv) * inv_eps;
        a4[j] = v;
        amax = fmaxf(amax, fmaxf(fmaxf(v.x, v.y), fmaxf(v.z, v.w)));
    }
    amax = wred_max(amax);
    float s = 0.f;
    #pragma unroll
    for (int j = 0; j < 4; j++) {
        s += expf(a4[j].x - amax) + expf(a4[j].y - amax)
           + expf(a4[j].z - amax) + expf(a4[j].w - amax);
    }
    s = wred_sum(s);
    // softmin = -eps * (logsumexp - log n)
    float res = -eps * (amax + logf(s) - LOG_N);

    if (finalMode) {
        __shared__ float wacc[8];
        if (lane == 0) {
            float sign = (mat < 2) ? 1.0f : -1.0f;     // +(f_ba,g_ab) -(f_aa,g_bb)
            wacc[wave] = sign * res * (1.0f / (NPER * 56.0f));
        }
        __syncthreads();
        if (threadIdx.x == 0) {
            float b = 0.f;
            #pragma unroll
            for (int w = 0; w < 8; w++) b += wacc[w];
            atomicAdd(out, b);                          // one atomic per block
        }
    } else if (lane == 0) {
        size_t o = ((size_t)pair * 4 + mat) * NPER + row;
        float oldv = (mixOld != 0.f) ? potIn[o] : 0.f;
        potOut[o] = mixOld * oldv + mixNew * res;
    }
}

__global__ void k_zero(float* __restrict__ p, int n) {
    int i = blockIdx.x * blockDim.x + threadIdx.x;
    if (i < n) p[i] = 0.f;
}

// ---------------------------------------------------------------------------
extern "C" void kernel_launch(void* const* d_in, const int* in_sizes, int n_in,
                              void* d_out, int out_size, void* d_ws, size_t ws_size,
                              hipStream_t stream) {
    const float* features = (const float*)d_in[0];
    float* out = (float*)d_out;
    float* ws  = (float*)d_ws;

    size_t off = 0;
    float* feats = ws + off; off += (size_t)NSELF * NPER * DIMF;   // 16 MB
    float* n2    = ws + off; off += 8192;
    float* eps0  = ws + off; off += 64;
    float* pot0  = ws + off; off += (size_t)NPAIRS * 4 * NPER;     // [pair][4][512]
    float* pot1  = ws + off; off += (size_t)NPAIRS * 4 * NPER;
    float* Cxy   = ws + off; off += (size_t)NPAIRS * MATSZ;        // 56 MB
    float* Cyx   = ws + off; off += (size_t)NPAIRS * MATSZ;        // 56 MB
    float* Cself = ws + off; off += (size_t)NSELF * MATSZ;         // 16 MB

    // setup
    k_gather<<<(NSELF * NPER * DIMF) / 256, 256, 0, stream>>>(features, feats);
    k_norms <<<1024, 256, 0, stream>>>(feats, n2);
    k_eps0  <<<NPAIRS, 256, 0, stream>>>(feats, eps0);
    k_gemm  <<<dim3(128, 128), 256, 0, stream>>>(feats, n2, Cxy, Cyx, Cself);

    int potN = 2 * NPAIRS * 4 * NPER;                  // pot0 & pot1 contiguous
    k_zero<<<(potN + 255) / 256, 256, 0, stream>>>(pot0, potN);
    k_zero<<<1, 32, 0, stream>>>(out, out_size);

    const int SM_GRID = NPAIRS * 4 * 64;               // 14336 blocks

    // init: f = softmin(eps0, C, 0); g read from zeroed pot1, write pot0
    k_softmin<<<SM_GRID, 256, 0, stream>>>(Cxy, Cyx, Cself, eps0,
                                           pot1, pot0,
                                           1.0f, 0.0f, 1.0f, 0, nullptr);
    // 140 annealed Jacobi steps, eps_i = max(eps0 * 0.81^i, 1e-8)
    float* bufs[2] = {pot0, pot1};
    double sc = 1.0;
    for (int i = 0; i < 140; i++) {
        k_softmin<<<SM_GRID, 256, 0, stream>>>(Cxy, Cyx, Cself, eps0,
                                               bufs[i & 1], bufs[(i + 1) & 1],
                                               (float)sc, 0.5f, 0.5f, 0, nullptr);
        sc *= 0.81;
    }
    // final extrapolation at eps = 1e-8, accumulate loss into d_out
    k_softmin<<<SM_GRID, 256, 0, stream>>>(Cxy, Cyx, Cself, eps0,
                                           bufs[0], nullptr,
                                           0.0f, 0.0f, 0.0f, 1, out);
}